// ViscRNNCellGen_63780264345628
// MI455X (gfx1250) — compile-verified
//
#include <hip/hip_runtime.h>
#include <stdint.h>

// Problem constants (match the reference).
#define B      128
#define T      4096
#define CIN    66            // [dt, sig0, tau(64)]
#define COUT   65            // [sig0, h(64)]
#define U      64            // units == threads per block (2 wave32s)
#define NC     32            // chunks along T
#define TC     (T / NC)      // 128 rows per chunk
#define TSUB   32            // rows per subtile (pipeline stage)
#define NSUB   (TC / TSUB)   // 4 subtiles per chunk
#define EPS    1e-5f
// exp(-dt/(2*tau_pos*1000)) = exp2(dt * rcp(tau_pos) * CEXP)
#define CEXP   (-7.2134752044448170368e-4f)   // -0.5 * log2(e) / 1000

#define SLOT_FLOATS (TSUB * CIN)        // 2112 floats
#define SLOT_BYTES  (SLOT_FLOATS * 4)   // 8448 bytes = 528 b128 packets
#define GROUP_ASYNC 9                   // async instructions per wave per subtile

// Typed pointers for the gfx1250 async-to-LDS builtin ("vV4i*1V4i*3IiIi").
typedef __attribute__((ext_vector_type(4))) int v4i;
typedef __attribute__((address_space(1))) v4i as1_v4i;
typedef __attribute__((address_space(3))) v4i as3_v4i;

// One async 16B global->LDS copy. OFF is the instruction immediate offset,
// which the hardware adds to BOTH the global and LDS addresses (ISA §10 async
// pseudocode), so a linear copy needs only one address pair per thread.
template <int OFF>
__device__ __forceinline__ void async_b128(const void* gsrc, void* lds_dst) {
#if __has_builtin(__builtin_amdgcn_global_load_async_to_lds_b128)
    __builtin_amdgcn_global_load_async_to_lds_b128(
        (as1_v4i*)(uintptr_t)gsrc,
        (as3_v4i*)(uint32_t)(uintptr_t)lds_dst, OFF, 0);
#else
    asm volatile("global_load_async_to_lds_b128 %0, %1, off offset:%2"
                 :: "v"((unsigned)(uintptr_t)lds_dst), "v"(gsrc), "i"(OFF)
                 : "memory");
#endif
}

template <int N>
__device__ __forceinline__ void async_wait() {
#if __has_builtin(__builtin_amdgcn_s_wait_asynccnt)
    __builtin_amdgcn_s_wait_asynccnt(N);
#else
    asm volatile("s_wait_asynccnt %0" :: "i"(N) : "memory");
#endif
}

// Issue one subtile (32 rows x 264B = 528 packets) as exactly 9 async
// instructions per wave: 8 full-width + 1 tail with 8 lanes active in EACH
// wave (keeps per-wave ASYNCcnt identical, so rolling waits are exact).
__device__ __forceinline__ void stage_sub(const char* sub_base, void* slot) {
    const int tid = threadIdx.x;
    const char* g0 = sub_base + tid * 16;
    char*       l0 = (char*)slot + tid * 16;
    async_b128<0 * 1024>(g0, l0);
    async_b128<1 * 1024>(g0, l0);
    async_b128<2 * 1024>(g0, l0);
    async_b128<3 * 1024>(g0, l0);
    async_b128<4 * 1024>(g0, l0);
    async_b128<5 * 1024>(g0, l0);
    async_b128<6 * 1024>(g0, l0);
    async_b128<7 * 1024>(g0, l0);
    const int lane = tid & 31;
    const int kp   = 512 + (tid >> 5) * 8 + lane;   // packets 512..527
    if (lane < 8)
        async_b128<0>(sub_base + kp * 16, (char*)slot + kp * 16);
}

// e = exp(-dt/(2*tau_pos*1000)); a = e*e; all native-rate (v_rcp/v_exp).
__device__ __forceinline__ float row_e(float dt, float tau) {
    float tp = fmaxf(tau, 0.0f) + EPS;
    return __builtin_amdgcn_exp2f(dt * __builtin_amdgcn_rcpf(tp) * CEXP);
}

// ---- Phase 1: local scan per chunk -> (prod(a), h_end) --------------------
__global__ void __launch_bounds__(U)
scan_local(const float* __restrict__ x, float* __restrict__ wsA,
           float* __restrict__ wsH) {
    __shared__ float tile[2][SLOT_FLOATS];
    const int b  = blockIdx.x >> 5;
    const int c  = blockIdx.x & (NC - 1);
    const int u  = threadIdx.x;
    const int t0 = c * TC;
    const char* cb = (const char*)(x + ((size_t)b * T + t0) * CIN);

    stage_sub(cb, tile[0]);
    stage_sub(cb + SLOT_BYTES, tile[1]);

    float sig_prev = (t0 == 0) ? 0.0f
                               : x[((size_t)b * T + t0 - 1) * CIN + 1];
    float A = 1.0f, h = 0.0f;

    #pragma unroll
    for (int s = 0; s < NSUB; ++s) {
        if (s == NSUB - 1) async_wait<0>(); else async_wait<GROUP_ASYNC>();
        __syncthreads();
        const float* bb = tile[s & 1];
        #pragma unroll 4
        for (int t = 0; t < TSUB; ++t) {
            const float* r = bb + t * CIN;
            float dt = r[0], sg = r[1], tau = r[2 + u];
            float e  = row_e(dt, tau);
            float a  = e * e;
            h = fmaf(a, h, e * (sg - sig_prev));
            A *= a;
            sig_prev = sg;
        }
        if (s + 2 < NSUB) {
            __syncthreads();  // all waves done reading this slot
            stage_sub(cb + (size_t)(s + 2) * SLOT_BYTES, tile[s & 1]);
        }
    }
    const size_t i = ((size_t)b * NC + c) * U + u;
    wsA[i] = A;
    wsH[i] = h;
}

// ---- Phase 2: exclusive prefix over chunk summaries -> carries ------------
__global__ void __launch_bounds__(U)
chunk_prefix(const float* __restrict__ wsA, const float* __restrict__ wsH,
             float* __restrict__ wsC) {
    const int b = blockIdx.x;
    const int u = threadIdx.x;
    float carry = 0.0f;
    #pragma unroll
    for (int c = 0; c < NC; ++c) {
        const size_t i = ((size_t)b * NC + c) * U + u;
        wsC[i] = carry;
        carry  = fmaf(wsA[i], carry, wsH[i]);
    }
}

// ---- Phase 3: seeded scan, write [sig0 | h] -------------------------------
__global__ void __launch_bounds__(U)
scan_final(const float* __restrict__ x, const float* __restrict__ wsC,
           float* __restrict__ out) {
    __shared__ float tile[2][SLOT_FLOATS];
    const int b  = blockIdx.x >> 5;
    const int c  = blockIdx.x & (NC - 1);
    const int u  = threadIdx.x;
    const int t0 = c * TC;
    const char* cb = (const char*)(x + ((size_t)b * T + t0) * CIN);

    stage_sub(cb, tile[0]);
    stage_sub(cb + SLOT_BYTES, tile[1]);

    float sig_prev = (t0 == 0) ? 0.0f
                               : x[((size_t)b * T + t0 - 1) * CIN + 1];
    float h = wsC[((size_t)b * NC + c) * U + u];   // chunk-start carry

    #pragma unroll
    for (int s = 0; s < NSUB; ++s) {
        if (s == NSUB - 1) async_wait<0>(); else async_wait<GROUP_ASYNC>();
        __syncthreads();
        const float* bb = tile[s & 1];
        #pragma unroll 2
        for (int t = 0; t < TSUB; ++t) {
            const float* r = bb + t * CIN;
            float dt = r[0], sg = r[1], tau = r[2 + u];
            float e  = row_e(dt, tau);
            float a  = e * e;
            h = fmaf(a, h, e * (sg - sig_prev));
            sig_prev = sg;

            const size_t ro = ((size_t)b * T + t0 + s * TSUB + t) * COUT;
            if (u == 0) out[ro] = sg;   // sig0 column
            out[ro + 1 + u] = h;        // 64 contiguous floats, coalesced
        }
        if (s + 2 < NSUB) {
            __syncthreads();
            stage_sub(cb + (size_t)(s + 2) * SLOT_BYTES, tile[s & 1]);
        }
    }
}

// ---- Launch ---------------------------------------------------------------
extern "C" void kernel_launch(void* const* d_in, const int* in_sizes, int n_in,
                              void* d_out, int out_size, void* d_ws, size_t ws_size,
                              hipStream_t stream) {
    (void)in_sizes; (void)n_in; (void)out_size; (void)ws_size;
    const float* x   = (const float*)d_in[0];
    float*       out = (float*)d_out;

    // Workspace: 3 arrays of B*NC*U floats (3 MB total).
    float* wsA = (float*)d_ws;
    float* wsH = wsA + (size_t)B * NC * U;
    float* wsC = wsH + (size_t)B * NC * U;

    scan_local  <<<B * NC, U, 0, stream>>>(x, wsA, wsH);
    chunk_prefix<<<B,      U, 0, stream>>>(wsA, wsH, wsC);
    scan_final  <<<B * NC, U, 0, stream>>>(x, wsC, out);
}